// OutputPPBlock_11940009083128
// MI455X (gfx1250) — compile-verified
//
#include <hip/hip_runtime.h>
#include <hip/hip_bf16.h>

typedef float v2f __attribute__((ext_vector_type(2)));
typedef float v8f __attribute__((ext_vector_type(8)));

#define EMB 128
#define OUT_EMB 256
#define NRAD 6
#define MAX_K 256
#define LDS_PAD 4   // float4-aligned stores, conflict-free wmma-operand reads

// ---------------------------------------------------------------------------
// Kernel 1: edge transform + scatter-sum onto nodes.
//   tmp[e,f] = m[e,f] * sum_r rbf[e,r] * W_rbf[f,r]
//   t[src[e], f] += tmp[e,f]      (non-returning f32 atomics; t is L2-resident)
// One wave per edge; lane handles 4 channels (32*4 = 128). W_rbf in LDS.
// This kernel is the HBM-bound stage: streams m (819 MB) + rbf once.
// ---------------------------------------------------------------------------
__global__ void edge_scatter_kernel(const float* __restrict__ m,
                                    const float* __restrict__ rbf,
                                    const int*   __restrict__ src,
                                    const float* __restrict__ W_rbf,
                                    float*       __restrict__ t,
                                    int E) {
    __shared__ float sW[EMB * NRAD];
    for (int i = threadIdx.x; i < EMB * NRAD; i += blockDim.x)
        sW[i] = W_rbf[i];
    __syncthreads();

    const int lane          = threadIdx.x & 31;
    const int waveInBlock   = threadIdx.x >> 5;
    const int wavesPerBlock = blockDim.x >> 5;
    const int stride        = gridDim.x * wavesPerBlock;

    for (int e = blockIdx.x * wavesPerBlock + waveInBlock; e < E; e += stride) {
        const float* r = rbf + (size_t)e * NRAD;   // broadcast load across lanes
        const float r0 = r[0], r1 = r[1], r2 = r[2], r3 = r[3], r4 = r[4], r5 = r[5];
        const int node = src[e];
        const int f0   = lane * 4;

        const float4 mm = *(const float4*)(m + (size_t)e * EMB + f0);

        float p[4];
#pragma unroll
        for (int j = 0; j < 4; ++j) {
            const float* w = &sW[(f0 + j) * NRAD];
            p[j] = r0 * w[0] + r1 * w[1] + r2 * w[2] +
                   r3 * w[3] + r4 * w[4] + r5 * w[5];
        }

        float* dst = t + (size_t)node * EMB + f0;
        unsafeAtomicAdd(dst + 0, mm.x * p[0]);
        unsafeAtomicAdd(dst + 1, mm.y * p[1]);
        unsafeAtomicAdd(dst + 2, mm.z * p[2]);
        unsafeAtomicAdd(dst + 3, mm.w * p[3]);
    }
}

// ---------------------------------------------------------------------------
// Kernel 2: WMMA f32 GEMM  Out[rows,256] = A[rows,K] x W[256,K]^T
//
// One 256-thread block (8 waves) per 16-row block. The 16xK A tile is staged
// into LDS once and shared by all 8 waves; each wave computes TWO 16x16
// column tiles (8 waves * 32 cols = 256 cols), amortizing the A fetch.
//
// V_WMMA_F32_16X16X4_F32 operand layout (ISA §7.12.2, 32-bit A 16x4):
//   lanes 0-15  : a = {A[M=lr][k+0], A[M=lr][k+1]}
//   lanes 16-31 : a = {A[M=lr][k+2], A[M=lr][k+3]}
//   B mirrors with N on lanes; C/D: vgpr v -> M=v (lanes 0-15), M=v+8 (16-31)
//
// mode: 0 = plain store; 1 = bias+SiLU store;
//       2 = bias+SiLU then atomic scatter into gout[gids[row]] (graph readout)
// ---------------------------------------------------------------------------
__global__ void gemm_wmma_kernel(const float* __restrict__ A,     // [rows, K]
                                 const float* __restrict__ W,     // [256, K]
                                 const float* __restrict__ bias,  // [256]
                                 float*       __restrict__ Out,   // [rows, 256]
                                 const int*   __restrict__ gids,  // [rows] (mode 2)
                                 float*       __restrict__ gout,  // [G, 256] (mode 2)
                                 int K, int mode) {
    __shared__ float sA[16 * (MAX_K + LDS_PAD)];
    const int strideA = K + LDS_PAD;
    const int rowBase = blockIdx.x * 16;

    // ---- stage A tile [16 x K] into LDS (float4 chunks) ----
    const int totalA = 16 * K;
    for (int idx = threadIdx.x * 4; idx < totalA; idx += blockDim.x * 4) {
        const int r = idx / K;
        const int k = idx % K;
        const float4 v = *(const float4*)(A + (size_t)(rowBase + r) * K + k);
        *(float4*)&sA[r * strideA + k] = v;
    }
    __syncthreads();

    const int lane = threadIdx.x & 31;
    const int wave = threadIdx.x >> 5;      // 0..7
    const int half = lane >> 4;             // 0: K+{0,1}, 1: K+{2,3}
    const int lr   = lane & 15;

    const int colBase0 = wave * 32;         // first 16-col tile for this wave
    const int colBase1 = colBase0 + 16;     // second

    v8f c0 = {0.f, 0.f, 0.f, 0.f, 0.f, 0.f, 0.f, 0.f};
    v8f c1 = {0.f, 0.f, 0.f, 0.f, 0.f, 0.f, 0.f, 0.f};

    const float* sAp   = &sA[lr * strideA + 2 * half];
    const float* Wptr0 = W + (size_t)(colBase0 + lr) * K + 2 * half;
    const float* Wptr1 = W + (size_t)(colBase1 + lr) * K + 2 * half;

#pragma unroll 4
    for (int k = 0; k < K; k += 4) {
        v2f a  = *(const v2f*)(sAp + k);       // ds_load_b64, conflict-free
        v2f b0 = *(const v2f*)(Wptr0 + k);
        v2f b1 = *(const v2f*)(Wptr1 + k);
        c0 = __builtin_amdgcn_wmma_f32_16x16x4_f32(false, a, false, b0,
                                                   (short)0, c0, false, false);
        c1 = __builtin_amdgcn_wmma_f32_16x16x4_f32(false, a, false, b1,
                                                   (short)0, c1, false, false);
    }

    // ---- epilogue ----
    const int col0 = colBase0 + lr;
    const int col1 = colBase1 + lr;
    const float b0 = (mode != 0) ? bias[col0] : 0.0f;
    const float b1 = (mode != 0) ? bias[col1] : 0.0f;

#pragma unroll
    for (int v = 0; v < 8; ++v) {
        const int row = rowBase + v + 8 * half;
        float x0 = c0[v];
        float x1 = c1[v];
        if (mode != 0) {
            x0 += b0;  x0 = x0 / (1.0f + __expf(-x0));   // SiLU
            x1 += b1;  x1 = x1 / (1.0f + __expf(-x1));
        }
        if (mode == 2) {
            const int g = gids[row];                      // broadcast within halves
            unsafeAtomicAdd(gout + (size_t)g * OUT_EMB + col0, x0);
            unsafeAtomicAdd(gout + (size_t)g * OUT_EMB + col1, x1);
        } else {
            Out[(size_t)row * OUT_EMB + col0] = x0;
            Out[(size_t)row * OUT_EMB + col1] = x1;
        }
    }
}

extern "C" void kernel_launch(void* const* d_in, const int* in_sizes, int n_in,
                              void* d_out, int out_size, void* d_ws, size_t ws_size,
                              hipStream_t stream) {
    const float* m       = (const float*)d_in[0];
    const float* rbf     = (const float*)d_in[1];
    const int*   src     = (const int*)  d_in[2];
    const int*   gids    = (const int*)  d_in[3];
    const float* W_rbf   = (const float*)d_in[4];
    const float* W_up    = (const float*)d_in[5];  // [256,128]
    const float* W_dense = (const float*)d_in[6];  // [3,256,256]
    const float* b_dense = (const float*)d_in[7];  // [3,256]

    const int E = in_sizes[2];   // num edges
    const int N = in_sizes[3];   // num nodes (50000 = 3125 * 16)

    float* t    = (float*)d_ws;                       // [N, 128]
    float* bufA = t    + (size_t)N * EMB;             // [N, 256]
    float* bufB = bufA + (size_t)N * OUT_EMB;         // [N, 256]

    // zero accumulators (graph-capture safe)
    hipMemsetAsync(t, 0, (size_t)N * EMB * sizeof(float), stream);
    hipMemsetAsync(d_out, 0, (size_t)out_size * sizeof(float), stream);

    // 1) edge transform + scatter to nodes (HBM-bound stage)
    {
        const int block = 256;                         // 8 waves
        const int wavesPerBlock = block / 32;
        const int grid = (E + wavesPerBlock - 1) / wavesPerBlock;
        edge_scatter_kernel<<<grid, block, 0, stream>>>(m, rbf, src, W_rbf, t, E);
    }

    const int rowBlocks = N / 16;   // 3125

    // 2) up-projection: bufA = t @ W_up^T   [N,128] -> [N,256]
    gemm_wmma_kernel<<<rowBlocks, 256, 0, stream>>>(
        t, W_up, b_dense, bufA, gids, (float*)d_out, EMB, /*mode=*/0);

    // 3) dense layers 0,1 with bias + SiLU (ping-pong)
    gemm_wmma_kernel<<<rowBlocks, 256, 0, stream>>>(
        bufA, W_dense + 0 * OUT_EMB * OUT_EMB, b_dense + 0 * OUT_EMB, bufB,
        gids, (float*)d_out, OUT_EMB, /*mode=*/1);
    gemm_wmma_kernel<<<rowBlocks, 256, 0, stream>>>(
        bufB, W_dense + 1 * OUT_EMB * OUT_EMB, b_dense + 1 * OUT_EMB, bufA,
        gids, (float*)d_out, OUT_EMB, /*mode=*/1);

    // 4) dense layer 2 fused with per-graph extensive readout (atomic scatter)
    gemm_wmma_kernel<<<rowBlocks, 256, 0, stream>>>(
        bufA, W_dense + 2 * OUT_EMB * OUT_EMB, b_dense + 2 * OUT_EMB, bufB,
        gids, (float*)d_out, OUT_EMB, /*mode=*/2);
}